// DynamicGaussianModel_37726992728446
// MI455X (gfx1250) — compile-verified
//
#include <hip/hip_runtime.h>
#include <math.h>

// N points x M nodes, rank-3 distance matrix via WMMA F32 16x16x4,
// per-row top-16 smallest, global-mean-scaled softmax, gather + weighted sum.
#define KNEAR   16
#define EPSF    1e-6f
#define MAXM    2048   // node table cap (problem has M=2048); staged in LDS
#define TSTRIDE 20     // tile row stride in words: 16B-aligned rows, no bank conflicts

typedef __attribute__((ext_vector_type(2))) float v2f;
typedef __attribute__((ext_vector_type(8))) float v8f;

// ---------------------------------------------------------------------------
// Pass 1: one wave per 16-row block of points. Node positions + squared norms
// staged once per block in LDS (24KB + 8KB of the 320KB WGP LDS). WMMA gives
// the 16x16 Gram tile (K=3 padded to 4); squared distances via
// |a|^2+|b|^2-2ab; each lane owns one row's running top-16 (replace-max) in
// VGPRs; sqrt deferred to the 16 survivors (monotonic => same selection).
// ---------------------------------------------------------------------------
__global__ __launch_bounds__(256) void pass1_topk(
    const float* __restrict__ means,
    const float* __restrict__ nodes,
    float* __restrict__ rowSum,   // [N] per-row sum of selected distances
    float* __restrict__ vals,     // [N*16] selected distances (sqrt'ed, clamped)
    int*   __restrict__ inds,     // [N*16]
    int N, int M)
{
    __shared__ __align__(16) float nodesLds[MAXM * 3];     // 24 KB
    __shared__ float colNormLds[MAXM];                     //  8 KB
    __shared__ __align__(16) float ldsTile[8 * 16 * TSTRIDE]; // 10 KB

    const int tid       = threadIdx.x;
    const int lane      = tid & 31;
    const int waveInBlk = tid >> 5;
    const int rowBase   = (blockIdx.x * 8 + waveInBlk) * 16;
    const int half      = lane >> 4;             // 0: K=0,1  1: K=2,3
    const int l15       = lane & 15;
    float* myTile = &ldsTile[waveInBlk * 16 * TSTRIDE];

    const int Mc = (M < MAXM) ? M : MAXM;

    // ---- stage node table into LDS (vectorized), then per-node |b|^2
    {
        const int nF  = Mc * 3;
        const int nV4 = nF >> 2;
        const float4* src4 = (const float4*)nodes;
        float4* dst4 = (float4*)nodesLds;
        for (int i = tid; i < nV4; i += 256) dst4[i] = src4[i];
        for (int i = nV4 * 4 + tid; i < nF; i += 256) nodesLds[i] = nodes[i];
    }
    __syncthreads();
    for (int i = tid; i < Mc; i += 256) {
        float x = nodesLds[3 * i], y = nodesLds[3 * i + 1], z = nodesLds[3 * i + 2];
        colNormLds[i] = x * x + y * y + z * z;
    }
    __syncthreads();

    // ---- A fragment: 16 points x (x,y,z,0). Lane L<16 -> (x,y); L>=16 -> (z,0)
    int p  = rowBase + l15;
    int pc = p < N ? p : (N - 1);
    __builtin_prefetch(&means[3 * pc], 0, 3);              // global_prefetch_b8
    float px = means[3 * pc + 0];
    float py = means[3 * pc + 1];
    float pz = means[3 * pc + 2];
    v2f a;
    a.x = half ? pz : px;
    a.y = half ? 0.0f : py;
    float rowNormSq = px * px + py * py + pz * pz;         // lanes 0-15: rows 0-15

    // ---- hoist loop-invariant row-norm broadcasts (8 bpermutes, once)
    float rnArr[8];
#pragma unroll
    for (int r = 0; r < 8; ++r)
        rnArr[r] = __shfl(rowNormSq, r + half * 8);

    // ---- top-16 state on SQUARED distance (lane owns row l15; halves
    //      duplicate work so the wave stays convergent at barriers)
    float tv[KNEAR];
    int   ti[KNEAR];
#pragma unroll
    for (int s = 0; s < KNEAR; ++s) { tv[s] = 3.4e38f; ti[s] = 0; }
    float curMax  = 3.4e38f;
    int   curSlot = 0;

    const int ntiles = (Mc + 15) >> 4;
    for (int t = 0; t < ntiles; ++t) {
        // ---- B fragment + col norm straight from LDS
        int j = t * 16 + l15;
        bool valid = (j < Mc);
        float nx = valid ? nodesLds[3 * j + 0] : 1e15f;
        float ny = valid ? nodesLds[3 * j + 1] : 0.0f;
        float nz = valid ? nodesLds[3 * j + 2] : 0.0f;
        float colNorm = valid ? colNormLds[j] : 1e30f;
        v2f b;
        b.x = half ? nz : nx;
        b.y = half ? 0.0f : ny;

        // ---- 16x16x4 F32 WMMA: full-EXEC here, no divergence
        v8f c = {};
        c = __builtin_amdgcn_wmma_f32_16x16x4_f32(
                /*neg_a=*/false, a, /*neg_b=*/false, b,
                /*c_mod=*/(short)0, c, /*reuse_a=*/false, /*reuse_b=*/false);

        // ---- squared distances, clamped at 0; stage tile in LDS.
        // C layout: VGPR r, lane L -> row r + 8*(L>=16), col L&15.
#pragma unroll
        for (int r = 0; r < 8; ++r) {
            int   mrow = r + half * 8;
            float d2   = fmaxf(rnArr[r] + colNorm - 2.0f * c[r], 0.0f);
            myTile[mrow * TSTRIDE + l15] = d2;
        }
        __syncthreads();

        // ---- per-lane selection over the 16 columns of row l15 (4x b128 reads)
        const float4* rowp = (const float4*)&myTile[l15 * TSTRIDE];
        float4 q0 = rowp[0], q1 = rowp[1], q2 = rowp[2], q3 = rowp[3];
        float cand[16] = { q0.x, q0.y, q0.z, q0.w,  q1.x, q1.y, q1.z, q1.w,
                           q2.x, q2.y, q2.z, q2.w,  q3.x, q3.y, q3.z, q3.w };
#pragma unroll
        for (int cc = 0; cc < 16; ++cc) {
            float d   = cand[cc];
            int   col = t * 16 + cc;
            bool  take = (col < Mc) && (d < curMax);
            if (take) {
#pragma unroll
                for (int s = 0; s < KNEAR; ++s) {
                    bool repl = (s == curSlot);
                    tv[s] = repl ? d   : tv[s];
                    ti[s] = repl ? col : ti[s];
                }
                curMax = tv[0]; curSlot = 0;
#pragma unroll
                for (int s = 1; s < KNEAR; ++s) {
                    if (tv[s] > curMax) { curMax = tv[s]; curSlot = s; }
                }
            }
        }
        __syncthreads();
    }

    // ---- emit: sqrt only the 16 survivors (single v_sqrt_f32 each), clamp eps
    if (half == 0 && p < N) {
        float s = 0.0f;
#pragma unroll
        for (int k = 0; k < KNEAR; ++k) {
            float d = fmaxf(__builtin_amdgcn_sqrtf(tv[k]), EPSF);
            vals[(size_t)p * KNEAR + k] = d;
            inds[(size_t)p * KNEAR + k] = ti[k];
            s += d;
        }
        rowSum[p] = s;
    }
}

// ---------------------------------------------------------------------------
// Pass 1b: deterministic global mean of all N*K selected distances.
// Fixed-order strided accumulation + LDS tree -> bit-stable across replays.
// ---------------------------------------------------------------------------
__global__ __launch_bounds__(256) void reduce_scale(
    const float* __restrict__ rowSum, float* __restrict__ scaleOut, int N)
{
    __shared__ float sm[256];
    float s = 0.0f;
    for (int i = threadIdx.x; i < N; i += 256) s += rowSum[i];
    sm[threadIdx.x] = s;
    __syncthreads();
    for (int off = 128; off > 0; off >>= 1) {
        if ((int)threadIdx.x < off) sm[threadIdx.x] += sm[threadIdx.x + off];
        __syncthreads();
    }
    if (threadIdx.x == 0)
        *scaleOut = sm[0] / (float)((size_t)N * KNEAR) + EPSF;
}

// ---------------------------------------------------------------------------
// Pass 2: softmax(-vals/scale) over K=16, gather node_offsets[time_index],
// weighted reduce, add means.
// ---------------------------------------------------------------------------
__global__ __launch_bounds__(256) void pass2_blend(
    const float* __restrict__ means,
    const float* __restrict__ node_offsets,   // [F, M, 3]
    const int*   __restrict__ timeIdx,
    const float* __restrict__ vals,
    const int*   __restrict__ inds,
    const float* __restrict__ scalePtr,
    float* __restrict__ out,
    int N, int M)
{
    int n = blockIdx.x * 256 + threadIdx.x;
    if (n >= N) return;

    float invScale = 1.0f / (*scalePtr);
    const float* off = node_offsets + (size_t)(*timeIdx) * (size_t)M * 3;

    float x[KNEAR];
    float mx = -3.4e38f;
#pragma unroll
    for (int k = 0; k < KNEAR; ++k) {
        x[k] = -vals[(size_t)n * KNEAR + k] * invScale;
        mx = fmaxf(mx, x[k]);
    }
    float w[KNEAR];
    float wsum = 0.0f;
#pragma unroll
    for (int k = 0; k < KNEAR; ++k) { w[k] = expf(x[k] - mx); wsum += w[k]; }
    float inv = 1.0f / wsum;

    float ax = 0.0f, ay = 0.0f, az = 0.0f;
#pragma unroll
    for (int k = 0; k < KNEAR; ++k) {
        int   idx = inds[(size_t)n * KNEAR + k];
        float ww  = w[k] * inv;
        ax += ww * off[3 * idx + 0];
        ay += ww * off[3 * idx + 1];
        az += ww * off[3 * idx + 2];
    }
    out[3 * n + 0] = means[3 * n + 0] + ax;
    out[3 * n + 1] = means[3 * n + 1] + ay;
    out[3 * n + 2] = means[3 * n + 2] + az;
}

// ---------------------------------------------------------------------------
extern "C" void kernel_launch(void* const* d_in, const int* in_sizes, int n_in,
                              void* d_out, int out_size, void* d_ws, size_t ws_size,
                              hipStream_t stream)
{
    const float* means        = (const float*)d_in[0];
    const float* nodes        = (const float*)d_in[1];
    const float* node_offsets = (const float*)d_in[2];
    const int*   timeIdx      = (const int*)d_in[3];

    const int N = in_sizes[0] / 3;
    const int M = in_sizes[1] / 3;

    // Workspace layout (floats): [0..15] scale scalar (+pad), [16..16+N) rowSum,
    // then N*16 vals, then N*16 inds. ~17.3 MB for N=131072.
    float* ws      = (float*)d_ws;
    float* scaleP  = ws;
    float* rowSum  = ws + 16;
    float* vals    = ws + 16 + N;
    int*   inds    = (int*)(ws + 16 + N + (size_t)N * KNEAR);

    const int nRowBlocks = (N + 15) / 16;          // one wave each
    const int blocks1    = (nRowBlocks + 7) / 8;   // 8 waves per block

    pass1_topk<<<blocks1, 256, 0, stream>>>(means, nodes, rowSum, vals, inds, N, M);
    reduce_scale<<<1, 256, 0, stream>>>(rowSum, scaleP, N);
    pass2_blend<<<(N + 255) / 256, 256, 0, stream>>>(
        means, node_offsets, timeIdx, vals, inds, scaleP, (float*)d_out, N, M);
}